// TaskHeadModel_71751723646993
// MI455X (gfx1250) — compile-verified
//
#include <hip/hip_runtime.h>

typedef __attribute__((ext_vector_type(2))) float v2f;
typedef __attribute__((ext_vector_type(8))) float v8f;

#define N_NODES 50000
#define E_EDGES 800000
#define DIM 128
#define TASK 64

// ---------------------------------------------------------------------------
// Degree / normalization kernels
// ---------------------------------------------------------------------------
__global__ __launch_bounds__(256) void init_deg_kernel(float* __restrict__ deg) {
    int i = blockIdx.x * blockDim.x + threadIdx.x;
    if (i < N_NODES) deg[i] = 1.0f;  // self-loop contributes 1 to every node
}

__global__ __launch_bounds__(256) void deg_count_kernel(const long long* __restrict__ dst,
                                                        float* __restrict__ deg) {
    int e = blockIdx.x * blockDim.x + threadIdx.x;
    if (e < E_EDGES) atomicAdd(&deg[(int)dst[e]], 1.0f);
}

__global__ __launch_bounds__(256) void deg_rsqrt_kernel(float* __restrict__ deg) {
    int i = blockIdx.x * blockDim.x + threadIdx.x;
    if (i < N_NODES) {
        float d = deg[i];
        deg[i] = (d > 0.0f) ? rsqrtf(d) : 0.0f;
    }
}

__global__ __launch_bounds__(256) void zero_kernel(float* __restrict__ p, long long n) {
    long long i = (long long)blockIdx.x * blockDim.x + threadIdx.x;
    if (i < n) p[i] = 0.0f;
}

// ---------------------------------------------------------------------------
// Weight packing: Wp[p][n] = (W[2p][n], W[2p+1][n]) so a WMMA B-fragment
// (two K-adjacent elements of one column) is a single global_load_b64.
// ---------------------------------------------------------------------------
template <int KDIM, int ODIM>
__global__ __launch_bounds__(256) void pack_w_kernel(const float* __restrict__ W,
                                                     v2f* __restrict__ Wp) {
    int i = blockIdx.x * blockDim.x + threadIdx.x;
    if (i < (KDIM / 2) * ODIM) {
        int p = i / ODIM;
        int n = i - p * ODIM;
        v2f v;
        v.x = W[(size_t)(2 * p) * ODIM + n];
        v.y = W[(size_t)(2 * p + 1) * ODIM + n];
        Wp[i] = v;
    }
}

// ---------------------------------------------------------------------------
// Edge aggregation: one wave per edge (incl. self-loops), 4 floats per lane.
// Edge id derived from block/wave id -> wave-uniform index & norm loads.
// msg = h[src] * (dinv[src]*dinv[dst]); agg[dst] += msg  (L2-resident atomics)
// ---------------------------------------------------------------------------
__global__ __launch_bounds__(256) void scatter_agg_kernel(const long long* __restrict__ src_idx,
                                                          const long long* __restrict__ dst_idx,
                                                          const float* __restrict__ dinv,
                                                          const float* __restrict__ h,
                                                          float* __restrict__ agg) {
    const long long ET = (long long)E_EDGES + N_NODES;
    long long edge = (long long)blockIdx.x * 8 + (threadIdx.x >> 5);  // wave-uniform
    int lane = (int)(threadIdx.x & 31);
    if (edge >= ET) return;

    int s, d;
    if (edge < E_EDGES) {
        s = (int)src_idx[edge];
        d = (int)dst_idx[edge];
    } else {  // self-loop
        s = d = (int)(edge - E_EDGES);
    }
    float norm = dinv[s] * dinv[d];

    const float4 v = *((const float4*)(h + (size_t)s * DIM) + lane);
    float* ap = agg + (size_t)d * DIM + lane * 4;
    atomicAdd(ap + 0, v.x * norm);
    atomicAdd(ap + 1, v.y * norm);
    atomicAdd(ap + 2, v.z * norm);
    atomicAdd(ap + 3, v.w * norm);
}

// ---------------------------------------------------------------------------
// fp32 WMMA GEMM: D[M,ODIM] = f(A[M,KDIM]) @ W[KDIM,ODIM] (+ obias)
//   AMODE: 0 = A used raw, 1 = relu(A + abias[k]) fused, 2 = A + abias[k]
// One wave computes a 16-row stripe across all ODIM columns (A reused NT x).
// A fragment (16x4 f32, 2 VGPRs): lane L holds row L&15, K = kbase+2*(L>>4)..+1
// B fragment: single b64 from packed Wp.  C/D: VGPR j -> row j+8*(L>>4), col L&15
// ---------------------------------------------------------------------------
template <int KDIM, int ODIM, int AMODE, bool OUT_BIAS>
__global__ __launch_bounds__(256) void gemm_wmma_kernel(const float* __restrict__ A,
                                                        const v2f* __restrict__ Wp,
                                                        const float* __restrict__ abias,
                                                        const float* __restrict__ obias,
                                                        float* __restrict__ D,
                                                        int M) {
    constexpr int NT = ODIM / 16;
    int wave = (int)((blockIdx.x * blockDim.x + threadIdx.x) >> 5);
    int lane = (int)(threadIdx.x & 31);
    int row0 = wave * 16;
    if (row0 >= M) return;

    int r = lane & 15;
    int khalf = lane >> 4;  // 0 or 1

    v8f acc[NT];
#pragma unroll
    for (int t = 0; t < NT; ++t)
        acc[t] = (v8f){0.f, 0.f, 0.f, 0.f, 0.f, 0.f, 0.f, 0.f};

    const float* arow = A + (size_t)(row0 + r) * KDIM;
#pragma unroll 4
    for (int k = 0; k < KDIM; k += 4) {
        int kb = k + 2 * khalf;
        v2f a = *(const v2f*)(arow + kb);
        if (AMODE != 0) {
            v2f bb = *(const v2f*)(abias + kb);
            a.x += bb.x;
            a.y += bb.y;
            if (AMODE == 1) {
                a.x = fmaxf(a.x, 0.0f);
                a.y = fmaxf(a.y, 0.0f);
            }
        }
        const v2f* wrow = Wp + (size_t)(kb >> 1) * ODIM;
#pragma unroll
        for (int t = 0; t < NT; ++t) {
            v2f b = wrow[t * 16 + r];
            acc[t] = __builtin_amdgcn_wmma_f32_16x16x4_f32(
                false, a, false, b, (short)0, acc[t], false, false);
        }
    }

#pragma unroll
    for (int t = 0; t < NT; ++t) {
        int col = t * 16 + r;
        float bv = OUT_BIAS ? obias[col] : 0.0f;
#pragma unroll
        for (int j = 0; j < 8; ++j) {
            int row = row0 + j + 8 * khalf;
            D[(size_t)row * ODIM + col] = acc[t][j] + bv;
        }
    }
}

// ---------------------------------------------------------------------------
// Launch
// ---------------------------------------------------------------------------
static inline int cdiv(long long a, long long b) { return (int)((a + b - 1) / b); }

extern "C" void kernel_launch(void* const* d_in, const int* in_sizes, int n_in,
                              void* d_out, int out_size, void* d_ws, size_t ws_size,
                              hipStream_t stream) {
    const float*     x   = (const float*)d_in[0];
    const long long* ei  = (const long long*)d_in[1];  // int64 per reference
    const float*     W1  = (const float*)d_in[2];
    const float*     b1  = (const float*)d_in[3];
    const float*     W2  = (const float*)d_in[4];
    const float*     b2  = (const float*)d_in[5];
    const float*     Wh  = (const float*)d_in[6];
    const float*     bh  = (const float*)d_in[7];
    float*           out = (float*)d_out;

    const long long* srcI = ei;            // edge_index[0]
    const long long* dstI = ei + E_EDGES;  // edge_index[1]

    // Workspace: dinv [N] | bufA [N*DIM] | bufB [N*DIM] | Wp1 | Wp2 | Wph
    float* dinv = (float*)d_ws;
    float* bufA = dinv + N_NODES;
    float* bufB = bufA + (size_t)N_NODES * DIM;
    v2f*   Wp1  = (v2f*)(bufB + (size_t)N_NODES * DIM);
    v2f*   Wp2  = Wp1 + (DIM / 2) * DIM;
    v2f*   Wph  = Wp2 + (DIM / 2) * DIM;

    const long long ND = (long long)N_NODES * DIM;
    const long long ET = (long long)E_EDGES + N_NODES;

    const int waves_gemm    = cdiv(N_NODES, 16);  // 3125 row tiles
    const int gemm_blocks   = cdiv((long long)waves_gemm * 32, 256);
    const int scatter_blocks = cdiv(ET, 8);       // 8 waves (edges) per block

    // --- weight packing (independent of everything else) ----------------
    pack_w_kernel<DIM, DIM><<<cdiv((DIM / 2) * DIM, 256), 256, 0, stream>>>(W1, Wp1);
    pack_w_kernel<DIM, DIM><<<cdiv((DIM / 2) * DIM, 256), 256, 0, stream>>>(W2, Wp2);
    pack_w_kernel<DIM, TASK><<<cdiv((DIM / 2) * TASK, 256), 256, 0, stream>>>(Wh, Wph);

    // --- normalization: deg -> dinv -------------------------------------
    init_deg_kernel<<<cdiv(N_NODES, 256), 256, 0, stream>>>(dinv);
    deg_count_kernel<<<cdiv(E_EDGES, 256), 256, 0, stream>>>(dstI, dinv);
    deg_rsqrt_kernel<<<cdiv(N_NODES, 256), 256, 0, stream>>>(dinv);

    // --- layer 1: bufB = agg(x @ W1)  (bias+relu deferred into GEMM2) ----
    gemm_wmma_kernel<DIM, DIM, 0, false><<<gemm_blocks, 256, 0, stream>>>(
        x, Wp1, nullptr, nullptr, bufA, N_NODES);
    zero_kernel<<<cdiv(ND, 256), 256, 0, stream>>>(bufB, ND);
    scatter_agg_kernel<<<scatter_blocks, 256, 0, stream>>>(srcI, dstI, dinv, bufA, bufB);

    // --- layer 2: bufB = agg(relu(bufB + b1) @ W2)  (b2 deferred) --------
    gemm_wmma_kernel<DIM, DIM, 1, false><<<gemm_blocks, 256, 0, stream>>>(
        bufB, Wp2, b1, nullptr, bufA, N_NODES);
    zero_kernel<<<cdiv(ND, 256), 256, 0, stream>>>(bufB, ND);
    scatter_agg_kernel<<<scatter_blocks, 256, 0, stream>>>(srcI, dstI, dinv, bufA, bufB);

    // --- head: out = (bufB + b2) @ Wh + bh -------------------------------
    gemm_wmma_kernel<DIM, TASK, 2, true><<<gemm_blocks, 256, 0, stream>>>(
        bufB, Wph, b2, bh, out, N_NODES);
}